// InverseFrequencyLoss_89575837925692
// MI455X (gfx1250) — compile-verified
//
#include <hip/hip_runtime.h>
#include <stdint.h>

// Problem constants (fixed by the reference): B=8, C=19, H=512, W=1024
#define C_CLASSES 19
#define HW        524288      // H*W = 512*1024 (elements per channel plane)
#define TILE_PIX  1024        // pixels per workgroup tile
#define THREADS   256

typedef __attribute__((ext_vector_type(4))) unsigned int v4u;
typedef __attribute__((ext_vector_type(8))) int          v8i;
typedef __attribute__((ext_vector_type(4))) int          v4i;

// ---------------------------------------------------------------- zero counts
__global__ void zero_counts_kernel(unsigned* __restrict__ counts) {
    if (threadIdx.x < C_CLASSES) counts[threadIdx.x] = 0u;
}

// ---------------------------------------------------------------- histogram
// 16 MB of int32 targets; each block: 256 threads x 4 int4 loads (coalesced),
// LDS histogram (integer atomics -> deterministic), one global add per class.
__global__ __launch_bounds__(THREADS) void hist_kernel(
    const int* __restrict__ tgt, unsigned* __restrict__ counts) {
    __shared__ unsigned h[C_CLASSES];
    const int tid = threadIdx.x;
    if (tid < C_CLASSES) h[tid] = 0u;
    __syncthreads();
    const int4* t4 = (const int4*)tgt;
    const int base = blockIdx.x * (THREADS * 4);
#pragma unroll
    for (int k = 0; k < 4; ++k) {
        const int4 v = t4[base + k * THREADS + tid];
        atomicAdd(&h[v.x], 1u);
        atomicAdd(&h[v.y], 1u);
        atomicAdd(&h[v.z], 1u);
        atomicAdd(&h[v.w], 1u);
    }
    __syncthreads();
    if (tid < C_CLASSES) atomicAdd(&counts[tid], h[tid]);
}

// ---------------------------------------------------------------- main pass
// One workgroup handles 1024 consecutive pixels of one image. The 19x1024 f32
// tile (rows strided by HW*4 bytes in memory) is DMA'd into LDS by the Tensor
// Data Mover (one 2D descriptor), then 8 waves compute log-softmax + weighted
// NLL from LDS and tree-reduce to one partial pair per block.
__global__ __launch_bounds__(THREADS) void loss_kernel(
    const float* __restrict__ x, const int* __restrict__ tgt,
    const unsigned* __restrict__ counts, float* __restrict__ partials) {
    __shared__ float tile[C_CLASSES * TILE_PIX];   // 76 KB: tile[c*1024 + p]
    __shared__ float invf[C_CLASSES];
    __shared__ float sA[THREADS];
    __shared__ float sB[THREADS];

    const int tid = threadIdx.x;
    const int blk = blockIdx.x;
    const int b   = blk >> 9;            // HW/TILE_PIX = 512 tiles per image
    const int p0  = (blk & 511) << 10;   // first pixel of tile within plane

    // inverse-frequency table (wave 0, before the barrier)
    if (tid < C_CLASSES) {
        const unsigned c  = counts[tid];
        const unsigned cm = (c < 1u) ? 1u : c;
        invf[tid] = (c > 0u) ? (1.0f / (float)cm) : 1.0f;
    }

    // TDM: wave 0 issues one 2D tensor load (19 rows x 1024 elems, row stride
    // HW elements) into LDS, waits TENSORcnt==0, then barrier releases all.
    if (tid < 32) {
        const uint64_t ga =
            (uint64_t)(uintptr_t)x +
            ((uint64_t)b * (uint64_t)C_CLASSES * (uint64_t)HW + (uint64_t)p0) * 4ull;
        const uint32_t lds = (uint32_t)(uintptr_t)&tile[0];

        v4u g0; v8i g1; v4i g2; v4i g3; v8i g4;
        // group0: [1:0]=count=1, [63:32]=lds_addr, [120:64]=global_addr, [127:126]=type=2
        g0[0] = 1u;
        g0[1] = lds;
        g0[2] = (uint32_t)ga;
        g0[3] = (uint32_t)(ga >> 32) | 0x80000000u;
        // group1: data_size=2 (4B); tensor_dim0=2^20 (no OOB), tensor_dim1=64;
        // tile_dim0=1024, tile_dim1=19, tile_dim2=0; tensor_dim0_stride=HW
        g1[0] = (int)(2u << 16);                       // data_size=4B, no mask/pad/iter
        g1[1] = 0;                                     // abar=0, tensor_dim0[15:0]=0
        g1[2] = (int)(0x0010u | (64u << 16));          // tensor_dim0[31:16]=0x10, tensor_dim1=64
        g1[3] = (int)((unsigned)TILE_PIX << 16);       // tensor_dim1 hi=0, tile_dim0=1024
        g1[4] = (int)C_CLASSES;                        // tile_dim1=19, tile_dim2=0
        g1[5] = (int)HW;                               // tensor_dim0_stride[31:0]
        g1[6] = 0;                                     // stride0 hi, stride1 lo
        g1[7] = 0;
        g2[0] = g2[1] = g2[2] = g2[3] = 0;             // 2D: groups 2/3 unused
        g3[0] = g3[1] = g3[2] = g3[3] = 0;
        g4[0] = g4[1] = g4[2] = g4[3] = 0;             // extra group (clang-23 6-arg
        g4[4] = g4[5] = g4[6] = g4[7] = 0;             // form): zero-filled
        __builtin_amdgcn_tensor_load_to_lds(g0, g1, g2, g3, g4, 0);
        __builtin_amdgcn_s_wait_tensorcnt(0);
    }
    __syncthreads();

    // 4 pixels per thread, straight from LDS (two passes: max, then sum-exp)
    const int pix = tid << 2;
    float m0, m1, m2, m3;
    {
        const float4 v = *(const float4*)&tile[pix];
        m0 = v.x; m1 = v.y; m2 = v.z; m3 = v.w;
    }
#pragma unroll
    for (int c = 1; c < C_CLASSES; ++c) {
        const float4 v = *(const float4*)&tile[c * TILE_PIX + pix];
        m0 = fmaxf(m0, v.x); m1 = fmaxf(m1, v.y);
        m2 = fmaxf(m2, v.z); m3 = fmaxf(m3, v.w);
    }
    float s0 = 0.f, s1 = 0.f, s2 = 0.f, s3 = 0.f;
#pragma unroll
    for (int c = 0; c < C_CLASSES; ++c) {
        const float4 v = *(const float4*)&tile[c * TILE_PIX + pix];
        s0 += __expf(v.x - m0); s1 += __expf(v.y - m1);
        s2 += __expf(v.z - m2); s3 += __expf(v.w - m3);
    }

    const int4 t4 = *(const int4*)&tgt[b * HW + p0 + pix];
    const float xt0 = tile[t4.x * TILE_PIX + pix + 0];
    const float xt1 = tile[t4.y * TILE_PIX + pix + 1];
    const float xt2 = tile[t4.z * TILE_PIX + pix + 2];
    const float xt3 = tile[t4.w * TILE_PIX + pix + 3];
    const float w0 = invf[t4.x], w1 = invf[t4.y], w2 = invf[t4.z], w3 = invf[t4.w];

    const float n0 = __logf(s0) + m0 - xt0;
    const float n1 = __logf(s1) + m1 - xt1;
    const float n2 = __logf(s2) + m2 - xt2;
    const float n3 = __logf(s3) + m3 - xt3;

    sA[tid] = w0 * n0 + w1 * n1 + w2 * n2 + w3 * n3;
    sB[tid] = w0 + w1 + w2 + w3;
    __syncthreads();
#pragma unroll
    for (int s = THREADS / 2; s > 0; s >>= 1) {
        if (tid < s) { sA[tid] += sA[tid + s]; sB[tid] += sB[tid + s]; }
        __syncthreads();
    }
    if (tid == 0) {
        partials[2 * blk + 0] = sA[0];
        partials[2 * blk + 1] = sB[0];
    }
}

// ---------------------------------------------------------------- finalize
__global__ __launch_bounds__(THREADS) void finalize_kernel(
    const float* __restrict__ partials, int nblk, float* __restrict__ out) {
    __shared__ float sA[THREADS];
    __shared__ float sB[THREADS];
    const int tid = threadIdx.x;
    float a = 0.f, b = 0.f;
    for (int i = tid; i < nblk; i += THREADS) {
        a += partials[2 * i + 0];
        b += partials[2 * i + 1];
    }
    sA[tid] = a; sB[tid] = b;
    __syncthreads();
#pragma unroll
    for (int s = THREADS / 2; s > 0; s >>= 1) {
        if (tid < s) { sA[tid] += sA[tid + s]; sB[tid] += sB[tid + s]; }
        __syncthreads();
    }
    if (tid == 0) out[0] = sA[0] / sB[0];
}

// ---------------------------------------------------------------- launch
extern "C" void kernel_launch(void* const* d_in, const int* in_sizes, int n_in,
                              void* d_out, int out_size, void* d_ws, size_t ws_size,
                              hipStream_t stream) {
    const float* x   = (const float*)d_in[0];   // [B,C,H,W] f32 logits
    const int*   tgt = (const int*)d_in[1];     // [B,H,W] int32 targets

    unsigned* counts   = (unsigned*)d_ws;                    // 19 x u32
    float*    partials = (float*)((char*)d_ws + 1024);       // nblk x float2

    const int npix     = in_sizes[1];            // 4,194,304
    const int nblk     = npix / TILE_PIX;        // 4096
    const int hist_blk = npix / (THREADS * 16);  // 1024 (16 ints per thread)

    zero_counts_kernel<<<1, 32, 0, stream>>>(counts);
    hist_kernel<<<hist_blk, THREADS, 0, stream>>>(tgt, counts);
    loss_kernel<<<nblk, THREADS, 0, stream>>>(x, tgt, counts, partials);
    finalize_kernel<<<1, THREADS, 0, stream>>>(partials, nblk, (float*)d_out);
}